// HashNGramEmbedding_80891414053418
// MI455X (gfx1250) — compile-verified
//
#include <hip/hip_runtime.h>
#include <stdint.h>

// HashNGramEmbedding for MI455X (gfx1250).
// Memory-bound gather: ~235 MB of traffic -> ~10us floor at 23.3 TB/s.
// CDNA5 path: GLOBAL_LOAD_ASYNC_TO_LDS_B128 (ASYNCcnt) in SADDR (GVS) form for
// the 512B row gathers; one wave32 owns PPW=2 output rows (lane l -> float4 at
// 16*l), so every random row gather is a fully-coalesced 512B burst. NT stores
// keep L2 for the embedding tables.

#define VOCAB_MASK      0x3FFFFu    // 2^18 - 1
#define EMBED_DIM       128
#define NTAB            6
#define SEQ             8192
#define BATCH           8
#define PPW             2           // positions per wave
#define WAVES_PER_BLOCK 8
#define THREADS         (WAVES_PER_BLOCK * 32)
#define POS_PER_BLOCK   (PPW * WAVES_PER_BLOCK)
#define TOTAL_POS       (BATCH * SEQ)
#define NBLOCKS         (TOTAL_POS / POS_PER_BLOCK)   // 4096

typedef __attribute__((ext_vector_type(4))) float v4f;  // native clang vector
                                                        // (accepted by nontemporal builtins)

__global__ __launch_bounds__(THREADS)
void hash_ngram_embed_kernel(const long long* __restrict__ x,
                             const float* __restrict__ emb,
                             float* __restrict__ out)
{
    // 8 waves * 2 positions * 6 tables * 512B = 48KB per block (320KB/WGP -> ~6 blocks/WGP)
    __shared__ __align__(16) float smem[WAVES_PER_BLOCK * PPW * NTAB * EMBED_DIM];

    const int tid  = threadIdx.x;
    const int wave = tid >> 5;
    const int lane = tid & 31;

    const int gw    = blockIdx.x * WAVES_PER_BLOCK + wave;  // global wave id
    const int pbase = gw * PPW;                             // first position of this wave
    const int row   = pbase >> 13;                          // / SEQ  (batch index)
    const int s0    = pbase & (SEQ - 1);                    // seq index of first position

    // x is int64 with values in [0,256); read the low dword only (little-endian).
    const int* xrow = (const int*)(x + ((size_t)row << 13));

    // Bytes for seq positions s0-7 .. s0+PPW-1. Branch-free padding: load from
    // a clamped (always-valid) address, then select 0 for idx<0 (matches the
    // reference's jnp.pad). Avoids divergent exec-mask branches per element.
    int arr[7 + PPW];
#pragma unroll
    for (int i = 0; i < 7 + PPW; ++i) {
        const int idx  = s0 - 7 + i;
        const int cidx = idx < 0 ? 0 : idx;      // clamped, always in-bounds
        const int v    = xrow[2 * cidx];         // unconditional load
        arr[i] = idx < 0 ? 0 : v;                // v_cndmask, no branch
    }

    // Rolling polynomial hashes, nested across n: h_n = h_{n-1}*257 + b_{s-(n-1)},
    // masked by 2^18-1 each step (== mod vocab). Fallback to b_s when s < n-1.
    uint32_t h[PPW][NTAB];
#pragma unroll
    for (int u = 0; u < PPW; ++u) {
        const int s = s0 + u;
        const uint32_t b0 = (uint32_t)arr[7 + u];
        uint32_t cur = b0;
#pragma unroll
        for (int j = 1; j <= 7; ++j) {
            cur = (cur * 257u + (uint32_t)arr[7 + u - j]) & VOCAB_MASK;
            if (j >= 2) h[u][j - 2] = (s >= j) ? cur : b0;   // n = j+1 in {3..8}
        }
    }

    // Async gather, SADDR (GVS) form: addr = SGPR64(emb) + VGPR_I32(voff).
    // Whole table block spans 6*2^18*512B = 805MB < 2^31, so a 32-bit offset
    // covers it. Lane l pulls 16B of each 512B row into LDS; 12 outstanding
    // async b128 gathers per wave, consumed in two halves (in-order completion
    // => asynccnt<=6 means u=0's rows landed).
    const uint32_t lane16 = (uint32_t)lane * 16u;
#pragma unroll
    for (int u = 0; u < PPW; ++u) {
#pragma unroll
        for (int k = 0; k < NTAB; ++k) {
            // byte offset into emb: (k*2^18 + h) * 512 + lane*16
            const uint32_t voff = ((uint32_t)k << 27) + (h[u][k] << 9) + lane16;
            const float* dst = &smem[(((wave * PPW) + u) * NTAB + k) * EMBED_DIM + lane * 4];
            const uint32_t lds_off = (uint32_t)(uintptr_t)dst;  // flat LDS ptr low 32b = LDS byte addr
            asm volatile("global_load_async_to_lds_b128 %0, %1, %2"
                         :: "v"(lds_off), "v"(voff), "s"(emb) : "memory");
        }
    }

#pragma unroll
    for (int u = 0; u < PPW; ++u) {
        if (u == 0) {
            asm volatile("s_wait_asynccnt 0x6" ::: "memory");  // first 6 gathers done
        } else {
            asm volatile("s_wait_asynccnt 0x0" ::: "memory");  // all done
        }
        v4f acc = (v4f)(0.0f);
#pragma unroll
        for (int k = 0; k < NTAB; ++k) {
            const v4f v = *(const v4f*)
                &smem[(((wave * PPW) + u) * NTAB + k) * EMBED_DIM + lane * 4];
            acc += v;
        }
        v4f r;
        r.x = acc.x / 6.0f; r.y = acc.y / 6.0f;   // exact division: matches the
        r.z = acc.z / 6.0f; r.w = acc.w / 6.0f;   // reference's f32 divide bit-for-bit
        __builtin_nontemporal_store(
            r, (v4f*)&out[((size_t)(pbase + u)) * EMBED_DIM + lane * 4]);
    }
}

extern "C" void kernel_launch(void* const* d_in, const int* in_sizes, int n_in,
                              void* d_out, int out_size, void* d_ws, size_t ws_size,
                              hipStream_t stream) {
    const long long* x   = (const long long*)d_in[0];  // (8, 8192) int64
    const float*     emb = (const float*)d_in[1];      // (6, 262144, 128) f32
    float*           out = (float*)d_out;              // (8, 8192, 128) f32
    (void)in_sizes; (void)n_in; (void)out_size; (void)d_ws; (void)ws_size;
    hash_ngram_embed_kernel<<<NBLOCKS, THREADS, 0, stream>>>(x, emb, out);
}